// GNNDI_3367254360365
// MI455X (gfx1250) — compile-verified
//
#include <hip/hip_runtime.h>
#include <math.h>

#define B_      16
#define C_      48
#define T_      96
#define P_      12
#define FF_     64
#define STEPS_  7
#define CT_     (C_ * T_)   // 4608
#define CP_     (C_ * P_)   // 576
#define BCP_    (B_ * CP_)  // 9216
#define NROW_   (B_ * C_)   // 768
#define EPS_    1e-5f
#define ALPHA_  0.5f
#define NTHR_   1024

typedef __attribute__((ext_vector_type(2))) float v2f;
typedef __attribute__((ext_vector_type(8))) float v8f;

__device__ __forceinline__ float gelu_exact(float v) {
    return 0.5f * v * (1.0f + erff(v * 0.70710678118654752f));
}
__device__ __forceinline__ float wred_max(float v) {
#pragma unroll
    for (int o = 16; o > 0; o >>= 1) v = fmaxf(v, __shfl_xor(v, o, 32));
    return v;
}
__device__ __forceinline__ float wred_sum(float v) {
#pragma unroll
    for (int o = 16; o > 0; o >>= 1) v += __shfl_xor(v, o, 32);
    return v;
}

// Single persistent workgroup: all recurrent state in LDS, 7 steps in-kernel.
__global__ void __launch_bounds__(NTHR_)
gnndi_scan_kernel(const float* __restrict__ x,
                  const float* __restrict__ g0, const float* __restrict__ b0,
                  const float* __restrict__ g1, const float* __restrict__ b1,
                  const float* __restrict__ g2, const float* __restrict__ b2,
                  const float* __restrict__ Wagg, const float* __restrict__ bagg,
                  const float* __restrict__ W1,  const float* __restrict__ bm1,
                  const float* __restrict__ W2,  const float* __restrict__ bm2,
                  const float* __restrict__ wmsg, const float* __restrict__ bmsg,
                  float* __restrict__ out, float* __restrict__ ws) {
    extern __shared__ float smem[];
    float* s_res  = smem;              // [B][C][P] residual state, 9216
    float* s_t    = smem + BCP_;       // [B][P][C] normalized t,   9216
    float* s_Wagg = smem + 2 * BCP_;   // 144
    float* s_W1a  = s_Wagg + 144;      // 64
    float* s_W1b  = s_W1a + FF_;       // 64
    float* s_bm1  = s_W1b + FF_;       // 64
    float* s_W2   = s_bm1 + FF_;       // 64

    const int tid  = threadIdx.x;
    const int wave = tid >> 5;
    const int lane = tid & 31;

    const float wmsgv = wmsg[0];
    const float bmsgv = bmsg[0];
    const float bm2v  = bm2[0];

    // ---- weights to LDS ----
    if (tid < 144) s_Wagg[tid] = Wagg[tid];
    if (tid < FF_) {
        s_W1a[tid] = W1[tid * 2 + 0];
        s_W1b[tid] = W1[tid * 2 + 1];
        s_bm1[tid] = bm1[tid];
        s_W2[tid]  = W2[tid];
    }

    // ---- Phase 0: xn = BN(x) over batch axis; seed state & first patch ----
    for (int f = tid; f < CT_; f += NTHR_) {
        float sum = 0.f, sq = 0.f;
#pragma unroll
        for (int b = 0; b < B_; b++) { float v = x[b * CT_ + f]; sum += v; sq += v * v; }
        float mu  = sum * (1.0f / B_);
        float var = sq * (1.0f / B_) - mu * mu;
        float sc  = rsqrtf(var + EPS_) * g0[f];
        float sh  = b0[f];
        int c = f / T_, tt = f % T_;
#pragma unroll
        for (int b = 0; b < B_; b++) {
            float v = (x[b * CT_ + f] - mu) * sc + sh;
            ws[b * CT_ + f] = v;                         // xn to scratch (windows)
            if (tt < P_) {
                out[b * CT_ + f] = v;                    // first patch of output
                s_res[b * CP_ + c * P_ + tt] = v;        // recurrent state seed
            }
        }
    }
    __threadfence_block();
    __syncthreads();

    // ---- Precompute WMMA B fragments for Wagg^T (uniform across waves) ----
    // B[k][n] = Wagg[n][k], 4x16 chunks; lane<16 holds K=k0,k0+1; lane>=16 K=k0+2,k0+3
    v2f bfrag[3];
#pragma unroll
    for (int kc = 0; kc < 3; kc++) {
        int k0 = kc * 4;
        int kk = (lane < 16) ? k0 : (k0 + 2);
        int n  = lane & 15;
        v2f bf;
        bf.x = (n < P_) ? s_Wagg[n * P_ + kk]     : 0.f;
        bf.y = (n < P_) ? s_Wagg[n * P_ + kk + 1] : 0.f;
        bfrag[kc] = bf;
    }

    for (int s = 0; s < STEPS_; s++) {
        // prefetch next input window (one line per (b,c) row)
        if (tid < NROW_) {
            int bb = tid / C_, cc = tid % C_;
            __builtin_prefetch(&ws[bb * CT_ + cc * T_ + P_ * (s + 1)], 0, 0);
        }

        // ---- Phase 1: in-place BN(prev) with g1,b1 (576 features x 16 batch) ----
        if (tid < CP_) {
            int base = tid;  // c*P_ + p
            float sum = 0.f, sq = 0.f;
#pragma unroll
            for (int b = 0; b < B_; b++) { float v = s_res[b * CP_ + base]; sum += v; sq += v * v; }
            float mu  = sum * (1.0f / B_);
            float var = sq * (1.0f / B_) - mu * mu;
            float sc  = rsqrtf(var + EPS_) * g1[tid];
            float sh  = b1[tid];
#pragma unroll
            for (int b = 0; b < B_; b++) {
                int idx = b * CP_ + base;
                s_res[idx] = (s_res[idx] - mu) * sc + sh;
            }
        }
        __syncthreads();

        // ---- Phase 2: WMMA f32 16x16x4 — (768x12) @ WaggT(12x12), gelu, +window ----
        for (int tile = wave; tile < NROW_ / 16; tile += 32) {
            int rowBase = tile * 16;
            v8f acc = {0.f, 0.f, 0.f, 0.f, 0.f, 0.f, 0.f, 0.f};
#pragma unroll
            for (int kc = 0; kc < 3; kc++) {
                int k0 = kc * 4;
                int m  = lane & 15;
                int kk = (lane < 16) ? k0 : (k0 + 2);
                v2f af;
                af.x = s_res[(rowBase + m) * P_ + kk];
                af.y = s_res[(rowBase + m) * P_ + kk + 1];
                acc = __builtin_amdgcn_wmma_f32_16x16x4_f32(
                    false, af, false, bfrag[kc], (short)0, acc, false, false);
            }
            int q = lane & 15;
#pragma unroll
            for (int r = 0; r < 8; r++) {
                int m   = (lane < 16) ? r : (r + 8);
                int row = rowBase + m;
                if (q < P_) {
                    float v  = gelu_exact(acc[r] + bagg[q]);
                    int bb = row / C_, cc = row % C_;
                    float xw = ws[bb * CT_ + cc * T_ + P_ * (s + 1) + q];
                    s_res[row * P_ + q] = v + xw;   // res
                }
            }
        }
        __syncthreads();

        // ---- Phase 3: t = BN(res) with g2,b2, transposed into s_t[b][p][c] ----
        if (tid < CP_) {
            int c = tid / P_, pp = tid % P_;
            int base = tid;
            float sum = 0.f, sq = 0.f;
#pragma unroll
            for (int b = 0; b < B_; b++) { float v = s_res[b * CP_ + base]; sum += v; sq += v * v; }
            float mu  = sum * (1.0f / B_);
            float var = sq * (1.0f / B_) - mu * mu;
            float sc  = rsqrtf(var + EPS_) * g2[tid];
            float sh  = b2[tid];
#pragma unroll
            for (int b = 0; b < B_; b++) {
                s_t[b * CP_ + pp * C_ + c] = (s_res[b * CP_ + base] - mu) * sc + sh;
            }
        }
        __syncthreads();

        // ---- Phase 4: per-(b,p) edge MLP + top-3 + softmax + message passing ----
        // one wave per graph; lane j covers column j and j+32 (j+32<48 only for lane<16)
        for (int bp = wave; bp < B_ * P_; bp += 32) {
            int b = bp / P_, pp = bp % P_;
            const float* trow = &s_t[b * CP_ + pp * C_];
            float tj1 = trow[lane];
            bool  has2 = (lane + 32) < C_;
            float tj2 = has2 ? trow[lane + 32] : 0.f;
            float msg1 = tj1 * wmsgv + bmsgv;
            float msg2 = tj2 * wmsgv + bmsgv;
            for (int i = 0; i < C_; i++) {
                float ti = trow[i];   // LDS broadcast
                float e1 = bm2v, e2 = bm2v;
#pragma unroll 8
                for (int f = 0; f < FF_; f++) {
                    float w2f = s_W2[f];
                    float p1 = s_W1a[f] * ti + s_W1b[f] * tj1 + s_bm1[f];
                    float p2 = s_W1a[f] * ti + s_W1b[f] * tj2 + s_bm1[f];
                    e1 += w2f * gelu_exact(p1);
                    e2 += w2f * gelu_exact(p2);
                }
                if (!has2) e2 = -3.0e38f;
                // top-3 threshold via three masked wave-max reductions
                float a1 = e1, a2 = e2;
                float m1 = wred_max(fmaxf(a1, a2));
                a1 = (a1 == m1) ? -3.0e38f : a1;  a2 = (a2 == m1) ? -3.0e38f : a2;
                float m2 = wred_max(fmaxf(a1, a2));
                a1 = (a1 == m2) ? -3.0e38f : a1;  a2 = (a2 == m2) ? -3.0e38f : a2;
                float m3 = wred_max(fmaxf(a1, a2));
                bool k1 = (e1 >= m3);
                bool k2 = has2 && (e2 >= m3);
                float x1 = k1 ? __expf(e1 - m1) : 0.f;
                float x2 = k2 ? __expf(e2 - m1) : 0.f;
                float denom = wred_sum(x1 + x2);
                float num   = wred_sum(x1 * msg1 + x2 * msg2);
                if (lane == 0) {
                    s_res[b * CP_ + i * P_ + pp] += ALPHA_ * (num / denom);
                }
            }
        }
        __syncthreads();

        // ---- Phase 5: emit this step's 12 output columns ----
        for (int idx = tid; idx < BCP_; idx += NTHR_) {
            int b = idx / CP_, r = idx % CP_;
            int c = r / P_, pp = r % P_;
            out[b * CT_ + c * T_ + P_ * (s + 1) + pp] = s_res[idx];
        }
        __syncthreads();
    }
}

extern "C" void kernel_launch(void* const* d_in, const int* in_sizes, int n_in,
                              void* d_out, int out_size, void* d_ws, size_t ws_size,
                              hipStream_t stream) {
    (void)in_sizes; (void)n_in; (void)out_size; (void)ws_size;
    const float* x    = (const float*)d_in[0];
    const float* g0   = (const float*)d_in[1];
    const float* b0   = (const float*)d_in[2];
    const float* g1   = (const float*)d_in[3];
    const float* b1   = (const float*)d_in[4];
    const float* g2   = (const float*)d_in[5];
    const float* b2   = (const float*)d_in[6];
    const float* Wagg = (const float*)d_in[7];
    const float* bagg = (const float*)d_in[8];
    const float* W1   = (const float*)d_in[9];
    const float* bm1  = (const float*)d_in[10];
    const float* W2   = (const float*)d_in[11];
    const float* bm2  = (const float*)d_in[12];
    const float* wmsg = (const float*)d_in[13];
    const float* bmsg = (const float*)d_in[14];
    float* out = (float*)d_out;
    float* ws  = (float*)d_ws;   // B*C*T floats of xn scratch (294,912 B)

    size_t smemBytes = (size_t)(2 * BCP_ + 144 + 4 * FF_) * sizeof(float); // ~75 KB
    gnndi_scan_kernel<<<dim3(1), dim3(NTHR_), smemBytes, stream>>>(
        x, g0, b0, g1, b1, g2, b2, Wagg, bagg, W1, bm1, W2, bm2, wmsg, bmsg, out, ws);
}